// GRU_30777735643267
// MI455X (gfx1250) — compile-verified
//
#include <hip/hip_runtime.h>
#include <math.h>

typedef __attribute__((ext_vector_type(2))) float v2f;
typedef __attribute__((ext_vector_type(8))) float v8f;

#define B_   64
#define T_   512
#define I_   256
#define H_   512
#define G3H  1536   // 3*H

// ---------------------------------------------------------------------------
// Kernel 1: gi[B*T, 3H] = x[B*T, I] @ W_ih^T[I, 3H] + b_ih
// One wave per 16x16 output tile; K-loop of V_WMMA_F32_16X16X4_F32.
// A frag (16x4 f32): lane<16 -> M=lane,   {K=kb,kb+1}
//                    lane>=16-> M=lane-16,{K=kb+2,kb+3}   (contiguous float2)
// B frag (4x16 f32): same striping over W_ih rows (B[k][n] = W_ih[n][k]).
// ---------------------------------------------------------------------------
__global__ __launch_bounds__(256) void gru_gi_gemm(
    const float* __restrict__ x,    // [B*T, I]
    const float* __restrict__ Wih,  // [3H, I]
    const float* __restrict__ bih,  // [3H]
    float* __restrict__ gi)         // [B*T, 3H]
{
    const int lane = threadIdx.x & 31;
    const int wid  = threadIdx.x >> 5;
    const int w    = blockIdx.x * 8 + wid;     // global wave id
    const int NT   = G3H / 16;                 // 96 N-tiles
    const int mt   = w / NT;
    const int nt   = w % NT;
    const int mr   = lane & 15;
    const int half = lane >> 4;

    const float* arow = x   + (size_t)(mt * 16 + mr) * I_;
    const float* brow = Wih + (size_t)(nt * 16 + mr) * I_;

    v8f acc = {};
    #pragma unroll 8
    for (int kb = 0; kb < I_; kb += 4) {
        const int kc = kb + 2 * half;
        v2f a = *(const v2f*)(arow + kc);
        v2f b = *(const v2f*)(brow + kc);
        acc = __builtin_amdgcn_wmma_f32_16x16x4_f32(
            false, a, false, b, (short)0, acc, false, false);
    }

    const int   n    = nt * 16 + mr;
    const float bias = bih[n];
    // D layout: VGPR v, lanes0-15 -> M=v, lanes16-31 -> M=v+8; N = mr
    float* drow = gi + (size_t)(mt * 16 + 8 * half) * G3H + n;
    #pragma unroll
    for (int v = 0; v < 8; ++v)
        drow[(size_t)v * G3H] = acc[v] + bias;
}

// ---------------------------------------------------------------------------
// Kernel 2: one GRU timestep. Each wave owns a 16x16 tile of h_new[64,512]
// and privately reduces r/z/n gate tiles over K=H with 3 WMMAs per k-step
// (A fragment of h_prev reused across the three gates). W_hh (3MB) and
// h_prev (128KB) stay L2-resident across the 512 steps.
// ---------------------------------------------------------------------------
__global__ __launch_bounds__(256) void gru_step(
    const float* __restrict__ gi,     // [B*T, 3H] (b_ih already added)
    const float* __restrict__ Whh,    // [3H, H]
    const float* __restrict__ bhh,    // [3H]
    const float* __restrict__ hprev,  // [B, H]
    float* __restrict__ hnext,        // [B, H]
    float* __restrict__ out,          // [B, T, H]
    int t)
{
    const int lane = threadIdx.x & 31;
    const int wid  = threadIdx.x >> 5;
    const int w    = blockIdx.x * 8 + wid;     // 0..127
    const int NT   = H_ / 16;                  // 32 N-tiles
    const int mt   = w / NT;                   // 0..3  (batch tiles)
    const int nt   = w % NT;
    const int mr   = lane & 15;
    const int half = lane >> 4;

    const int n = nt * 16 + mr;                // H column this lane owns (B/C/D N)
    const float* arow = hprev + (size_t)(mt * 16 + mr) * H_;
    const float* wr   = Whh + (size_t)(0 * H_ + n) * H_;
    const float* wz   = Whh + (size_t)(1 * H_ + n) * H_;
    const float* wn   = Whh + (size_t)(2 * H_ + n) * H_;

    v8f accr = {}, accz = {}, accn = {};
    #pragma unroll 4
    for (int kb = 0; kb < H_; kb += 4) {
        const int kc = kb + 2 * half;
        v2f a  = *(const v2f*)(arow + kc);
        v2f br = *(const v2f*)(wr + kc);
        v2f bz = *(const v2f*)(wz + kc);
        v2f bn = *(const v2f*)(wn + kc);
        accr = __builtin_amdgcn_wmma_f32_16x16x4_f32(
            false, a, false, br, (short)0, accr, false, false);
        accz = __builtin_amdgcn_wmma_f32_16x16x4_f32(
            false, a, false, bz, (short)0, accz, false, false);
        accn = __builtin_amdgcn_wmma_f32_16x16x4_f32(
            false, a, false, bn, (short)0, accn, false, false);
    }

    const float bhr = bhh[n];
    const float bhz = bhh[H_ + n];
    const float bhn = bhh[2 * H_ + n];

    #pragma unroll
    for (int v = 0; v < 8; ++v) {
        const int    m     = mt * 16 + v + 8 * half;       // batch row
        const size_t girow = ((size_t)m * T_ + t) * G3H;
        const float ir = gi[girow + n];
        const float iz = gi[girow + H_ + n];
        const float in_ = gi[girow + 2 * H_ + n];
        const float hp = hprev[(size_t)m * H_ + n];

        const float r  = 1.0f / (1.0f + __expf(-(ir + accr[v] + bhr)));
        const float z  = 1.0f / (1.0f + __expf(-(iz + accz[v] + bhz)));
        const float ng = tanhf(in_ + r * (accn[v] + bhn));
        const float hn = (1.0f - z) * ng + z * hp;

        hnext[(size_t)m * H_ + n] = hn;
        out[(size_t)m * T_ * H_ + (size_t)t * H_ + n] = hn;
    }
}

__global__ void gru_zero(float* __restrict__ p, int nElem) {
    int i = blockIdx.x * blockDim.x + threadIdx.x;
    if (i < nElem) p[i] = 0.0f;
}

// ---------------------------------------------------------------------------
extern "C" void kernel_launch(void* const* d_in, const int* in_sizes, int n_in,
                              void* d_out, int out_size, void* d_ws, size_t ws_size,
                              hipStream_t stream) {
    (void)in_sizes; (void)n_in; (void)out_size; (void)ws_size;
    const float* x   = (const float*)d_in[0];  // [B,T,I]
    const float* Wih = (const float*)d_in[1];  // [3H,I]
    const float* Whh = (const float*)d_in[2];  // [3H,H]
    const float* bih = (const float*)d_in[3];  // [3H]
    const float* bhh = (const float*)d_in[4];  // [3H]
    float* out = (float*)d_out;                // [B,T,H] then [1,B,H]

    float* gi    = (float*)d_ws;                       // B*T*3H floats (~201 MB)
    float* hbuf0 = gi + (size_t)B_ * T_ * G3H;         // [B,H]
    float* hbuf1 = hbuf0 + (size_t)B_ * H_;            // [B,H]
    float* bufs[2] = {hbuf0, hbuf1};

    // h0 = 0
    gru_zero<<<(B_ * H_ + 255) / 256, 256, 0, stream>>>(hbuf0, B_ * H_);

    // gi = x @ W_ih^T + b_ih : (B*T/16)*(3H/16) tiles, 8 waves/block
    const int tiles = (B_ * T_ / 16) * (G3H / 16);     // 196608
    gru_gi_gemm<<<tiles / 8, 256, 0, stream>>>(x, Wih, bih, gi);

    // sequential scan: 128 tiles/step, 16 blocks x 8 waves
    for (int t = 0; t < T_; ++t) {
        gru_step<<<16, 256, 0, stream>>>(gi, Whh, bhh,
                                         bufs[t & 1], bufs[(t + 1) & 1],
                                         out, t);
    }

    // h_n -> tail of d_out
    hipMemcpyAsync(out + (size_t)B_ * T_ * H_, bufs[T_ & 1],
                   (size_t)B_ * H_ * sizeof(float),
                   hipMemcpyDeviceToDevice, stream);
}